// UnifiedGraphQLSTM_65481071405885
// MI455X (gfx1250) — compile-verified
//
#include <hip/hip_runtime.h>
#include <math.h>

// ---------------------------------------------------------------------------
// Quantum-LSTM on MI455X (gfx1250, wave32).
//
// Phase 1 (WMMA f32 16x16x4): pre[n][16] = feat[n][:2048] @ Wx^T + b + theta
// Phase 2 (1 wave, serial):   LSTM scan, qlayer reduced to cosine products
// Phase 3 (WMMA f32 16x16x4): out = h @ Wfin^T + bfin   (K = HIDDEN = 4)
// ---------------------------------------------------------------------------

typedef float v2f __attribute__((ext_vector_type(2)));
typedef float v8f __attribute__((ext_vector_type(8)));

#define N_NODES 8192
#define FEAT    2048
#define WROW    2052   // D = FEAT + HIDDEN
#define OUTD    2048

// ---------------------------------------------------------------------------
// Phase 1: 16-node x 16-col tile per wave; K-loop 2048 in steps of 8
// (two WMMAs with independent accumulators to pipeline the XDL chain).
// A layout (16x4 f32): lanes 0-15 -> M=lane, v0/v1 = K{0,1}; lanes 16-31 K{2,3}.
// B layout (4x16 f32): N = lane%16, v0/v1 = K{0,1} (+2 for upper lane half).
// C/D layout: N = lane%16, row M = r + 8*(lane/16).
// ---------------------------------------------------------------------------
__global__ __launch_bounds__(32)
void qlstm_gates_wmma(const float* __restrict__ feat,
                      const float* __restrict__ Wf, const float* __restrict__ bf,
                      const float* __restrict__ Wi, const float* __restrict__ bi,
                      const float* __restrict__ Wg, const float* __restrict__ bg,
                      const float* __restrict__ Wo, const float* __restrict__ bo,
                      const float* __restrict__ thf, const float* __restrict__ thi,
                      const float* __restrict__ thg, const float* __restrict__ tho,
                      float* __restrict__ pre)
{
    const int lane = threadIdx.x;          // 0..31
    const int n0   = blockIdx.x * 16;      // node tile base
    const int col  = lane & 15;            // output column: gate*4 + qubit
    const int hi   = lane >> 4;            // K-pair selector
    const int g    = col >> 2;
    const int q    = col & 3;

    const float* W  = (g == 0) ? Wf  : (g == 1) ? Wi  : (g == 2) ? Wg  : Wo;
    const float* bb = (g == 0) ? bf  : (g == 1) ? bi  : (g == 2) ? bg  : bo;
    const float* th = (g == 0) ? thf : (g == 1) ? thi : (g == 2) ? thg : tho;

    // Fold bias + theta into the accumulator (value depends only on column).
    const float s = bb[q] + th[q];
    v8f acc0, acc1;
#pragma unroll
    for (int r = 0; r < 8; ++r) { acc0[r] = s; acc1[r] = 0.0f; }

    const float* arow = feat + (size_t)(n0 + col) * FEAT + 2 * hi; // A row base
    const float* wrow = W    + (size_t)q * WROW + 2 * hi;          // B row base

    for (int k = 0; k < FEAT; k += 8) {
        v2f a0 = *(const v2f*)(arow + k);
        v2f b0 = *(const v2f*)(wrow + k);
        v2f a1 = *(const v2f*)(arow + k + 4);
        v2f b1 = *(const v2f*)(wrow + k + 4);
        acc0 = __builtin_amdgcn_wmma_f32_16x16x4_f32(false, a0, false, b0,
                                                     (short)0, acc0, false, false);
        acc1 = __builtin_amdgcn_wmma_f32_16x16x4_f32(false, a1, false, b1,
                                                     (short)0, acc1, false, false);
    }
#pragma unroll
    for (int r = 0; r < 8; ++r)
        pre[(size_t)(n0 + r + 8 * hi) * 16 + col] = acc0[r] + acc1[r];
}

// ---------------------------------------------------------------------------
// Phase 2: serial LSTM scan, one wave.
// qlayer(x,theta)[w] collapses to products of cos(phi_j), phi = pre + h@Wh^T:
//   z0=c1c2c3  z1=c0c1  z2=c0c1c2  z3=c0c1c2c3
// Lanes 0-15: (gate,qubit) = (lane/4, lane%4); lanes 16-31 mirror for full EXEC.
// ---------------------------------------------------------------------------
__global__ __launch_bounds__(32)
void qlstm_scan(const float* __restrict__ pre,  // [N_NODES][16]
                const float* __restrict__ Wf, const float* __restrict__ Wi,
                const float* __restrict__ Wg, const float* __restrict__ Wo,
                float* __restrict__ hout)       // [N_NODES][4]
{
    const int lane = threadIdx.x;
    const int role = lane & 15;
    const int g    = role >> 2;
    const int q    = role & 3;
    const int half = lane & 16;       // mirror base for shuffles
    const int grp  = lane & ~3;       // 4-lane gate group base

    const float* W = (g == 0) ? Wf : (g == 1) ? Wi : (g == 2) ? Wg : Wo;
    // Hidden-state weights Wh = W[q][2048:2052]
    const float wh0 = W[(size_t)q * WROW + FEAT + 0];
    const float wh1 = W[(size_t)q * WROW + FEAT + 1];
    const float wh2 = W[(size_t)q * WROW + FEAT + 2];
    const float wh3 = W[(size_t)q * WROW + FEAT + 3];

    float hx0 = 0.f, hx1 = 0.f, hx2 = 0.f, hx3 = 0.f;
    float cst = 0.f;                  // cell state for hidden index lane&3

    for (int n = 0; n < N_NODES; ++n) {
        float phi = pre[(size_t)n * 16 + role]
                  + wh0 * hx0 + wh1 * hx1 + wh2 * hx2 + wh3 * hx3;
        float cg = __cosf(phi);

        // Gather the gate group's four cosines.
        float c0 = __shfl(cg, grp + 0, 32);
        float c1 = __shfl(cg, grp + 1, 32);
        float c2 = __shfl(cg, grp + 2, 32);
        float c3 = __shfl(cg, grp + 3, 32);
        float p01 = c0 * c1;
        float z = (q == 0) ? c1 * c2 * c3
                : (q == 1) ? p01
                : (q == 2) ? p01 * c2
                :            p01 * c2 * c3;

        // f,i,o -> sigmoid ; g (gate index 2) -> tanh. |z|<=1 so fast exp is safe.
        float act;
        if (g == 2) {
            float e = __expf(2.0f * z);
            act = (e - 1.0f) / (e + 1.0f);
        } else {
            act = 1.0f / (1.0f + __expf(-z));
        }

        // Gather per-hidden-unit gate values (hidden index w = lane&3).
        float fv = __shfl(act, half + 0  + q, 32);
        float iv = __shfl(act, half + 4  + q, 32);
        float gv = __shfl(act, half + 8  + q, 32);
        float ov = __shfl(act, half + 12 + q, 32);

        cst = fv * cst + iv * gv;
        float ec = __expf(2.0f * cst);
        float hnew = ov * ((ec - 1.0f) / (ec + 1.0f));

        if (lane < 4) hout[(size_t)n * 4 + lane] = hnew;

        // Broadcast new hidden state to every lane.
        hx0 = __shfl(hnew, half + 0, 32);
        hx1 = __shfl(hnew, half + 1, 32);
        hx2 = __shfl(hnew, half + 2, 32);
        hx3 = __shfl(hnew, half + 3, 32);
    }
}

// ---------------------------------------------------------------------------
// Phase 3: out[n][o] = h[n][:] @ Wfin[o][:] + bfin[o].  K = 4 -> exactly one
// WMMA f32 16x16x4 per 16x16 tile. A is loaded once per node tile, B/bias per
// output tile (Wfin is 32 KB -> L2 resident).
// ---------------------------------------------------------------------------
__global__ __launch_bounds__(32)
void qlstm_out_wmma(const float* __restrict__ h,     // [N_NODES][4]
                    const float* __restrict__ Wfin,  // [OUTD][4]
                    const float* __restrict__ bfin,  // [OUTD]
                    float* __restrict__ out)         // [N_NODES][OUTD]
{
    const int lane = threadIdx.x;
    const int n0   = blockIdx.x * 16;
    const int nlo  = lane & 15;
    const int hi   = lane >> 4;

    // A tile: 16 nodes x K=4
    v2f a = *(const v2f*)(h + (size_t)(n0 + nlo) * 4 + 2 * hi);

    for (int o0 = 0; o0 < OUTD; o0 += 16) {
        v2f b = *(const v2f*)(Wfin + (size_t)(o0 + nlo) * 4 + 2 * hi);
        float bias = bfin[o0 + nlo];
        v8f acc;
#pragma unroll
        for (int r = 0; r < 8; ++r) acc[r] = bias;
        acc = __builtin_amdgcn_wmma_f32_16x16x4_f32(false, a, false, b,
                                                    (short)0, acc, false, false);
#pragma unroll
        for (int r = 0; r < 8; ++r)
            out[(size_t)(n0 + r + 8 * hi) * OUTD + o0 + nlo] = acc[r];
    }
}

// ---------------------------------------------------------------------------
extern "C" void kernel_launch(void* const* d_in, const int* in_sizes, int n_in,
                              void* d_out, int out_size, void* d_ws, size_t ws_size,
                              hipStream_t stream) {
    const float* feat = (const float*)d_in[0];
    const float* Wf   = (const float*)d_in[1];
    const float* bf   = (const float*)d_in[2];
    const float* Wi   = (const float*)d_in[3];
    const float* bi   = (const float*)d_in[4];
    const float* Wg   = (const float*)d_in[5];
    const float* bg   = (const float*)d_in[6];
    const float* Wo   = (const float*)d_in[7];
    const float* bo   = (const float*)d_in[8];
    const float* thf  = (const float*)d_in[9];
    const float* thi  = (const float*)d_in[10];
    const float* thg  = (const float*)d_in[11];
    const float* tho  = (const float*)d_in[12];
    const float* Wfin = (const float*)d_in[13];
    const float* bfin = (const float*)d_in[14];
    float* out = (float*)d_out;

    float* pre = (float*)d_ws;                 // 8192*16 floats = 512 KB
    float* h   = pre + (size_t)N_NODES * 16;   // 8192*4  floats = 128 KB

    qlstm_gates_wmma<<<N_NODES / 16, 32, 0, stream>>>(
        feat, Wf, bf, Wi, bi, Wg, bg, Wo, bo, thf, thi, thg, tho, pre);

    qlstm_scan<<<1, 32, 0, stream>>>(pre, Wf, Wi, Wg, Wo, h);

    qlstm_out_wmma<<<N_NODES / 16, 32, 0, stream>>>(h, Wfin, bfin, out);
}